// deformable_SKConv_48069273977419
// MI455X (gfx1250) — compile-verified
//
#include <hip/hip_runtime.h>
#include <hip/hip_bf16.h>

// ---------------------------------------------------------------------------
// Deformable SKConv, fully fused per 16-pixel tile, fp32 WMMA (16x16x4) on the
// CDNA5 matrix pipe. B=4, H=W=192, Cf=O=64, L=32, IN_NC=7.
// ---------------------------------------------------------------------------

#define BATCH 4
#define HH    192
#define WW    192
#define HWSZ  (HH * WW)         // 36864
#define CF    64
#define OC    64
#define LC    32
#define CIN   7

typedef __attribute__((ext_vector_type(2))) float v2f;
typedef __attribute__((ext_vector_type(8))) float v8f;

// D = A(16x4) * B(4x16) + C, fp32, one wave.
// A layout (32-bit, 16x4): lane&15 = M, VGPR v: K = v + 2*(lane>>4)
// B layout (32-bit, 4x16): lane&15 = N, VGPR v: K = v + 2*(lane>>4)
// C/D layout: VGPR r: M = r + 8*(lane>>4), N = lane&15
__device__ __forceinline__ v8f wmma4(v2f a, v2f b, v8f c) {
  return __builtin_amdgcn_wmma_f32_16x16x4_f32(
      /*neg_a=*/false, a, /*neg_b=*/false, b,
      /*c_mod=*/(short)0, c, /*reuse_a=*/false, /*reuse_b=*/false);
}

// ---------------------------------------------------------------------------
// Phase 1: per scale, fused dw-conv -> pw-conv -> deformable sampling -> dcn
// GEMM -> relu.  One wave per 16-pixel tile, 8 waves / block.
// ---------------------------------------------------------------------------
template <int KS>
__global__ __launch_bounds__(256) void scale_kernel(
    const float* __restrict__ fea, const float* __restrict__ inp,
    const float* __restrict__ dww, const float* __restrict__ dwb,
    const float* __restrict__ pww, const float* __restrict__ pwb,
    const float* __restrict__ dcnw, const float* __restrict__ dcnb,
    float* __restrict__ outw)   // [BATCH][64][HWSZ] for this scale
{
  constexpr int DS  = KS * KS;
  constexpr int CK  = CIN * DS;           // deformable K dimension
  constexpr int PAD = KS / 2;
  constexpr int NCH = (CK + 3) / 4;       // chunks of 4 taps

  __shared__ float dwbuf[8][CF * 16];     // [K=channel][M=pixel] per wave
  __shared__ float omch [8][16 * 16];     // [M=pixel][slot] per wave
  __shared__ float sampb[8][4 * 16];      // [K=tap][M=pixel] per wave

  const int wave = threadIdx.x >> 5;
  const int lane = threadIdx.x & 31;
  const int tile = blockIdx.x * 8 + wave;
  const int p0   = tile * 16;
  const int b    = p0 / HWSZ;
  const int rem  = p0 - b * HWSZ;
  const int y    = rem / WW;
  const int x0   = rem - y * WW;

  // ---- depthwise conv: 64 channels x 16 pixels, 32 values per lane --------
  for (int it = 0; it < 32; ++it) {
    const int lin = it * 32 + lane;
    const int c = lin >> 4, m = lin & 15;
    float v = dwb[c];
    const float* fp = fea + (size_t)(b * CF + c) * HWSZ;
#pragma unroll
    for (int dy = 0; dy < KS; ++dy) {
      const int yy = y + dy - PAD;
      if (yy < 0 || yy >= HH) continue;
#pragma unroll
      for (int dx = 0; dx < KS; ++dx) {
        const int xx = x0 + m + dx - PAD;
        if (xx < 0 || xx >= WW) continue;
        v += fp[yy * WW + xx] * dww[c * DS + dy * KS + dx];
      }
    }
    dwbuf[wave][c * 16 + m] = v;
  }
  __syncthreads();

  const int m    = lane & 15;     // A row / B col / D col
  const int half = lane >> 4;
  const int kb   = half * 2;      // base K of this lane's two regs

  v8f acc[4] = {};                // 16 pixels x 64 output channels

  for (int ch = 0; ch < NCH; ++ch) {
    // ---- pw-conv WMMA: compute the 12 om channels this chunk needs --------
    // slot n<12: tap j=n/3, role n%3 (0=off_y,1=off_x,2=mask)
    const int j   = m / 3;
    const int r3  = m - j * 3;
    const int pi0 = ch * 4 + j;
    const bool slotValid = (m < 12) && (pi0 < CK);
    int row = 0;
    if (slotValid) row = (r3 == 2) ? (2 * CIN * DS + pi0) : (2 * pi0 + r3);
    const float* wr = pww + row * CF;

    v8f d = {};
#pragma unroll
    for (int t = 0; t < CF / 4; ++t) {
      v2f a, bb;
      a.x  = dwbuf[wave][(4 * t + kb)     * 16 + m];
      a.y  = dwbuf[wave][(4 * t + kb + 1) * 16 + m];
      bb.x = wr[4 * t + kb];
      bb.y = wr[4 * t + kb + 1];
      d = wmma4(a, bb, d);
    }
    const float bias = slotValid ? pwb[row] : 0.0f;
#pragma unroll
    for (int r = 0; r < 8; ++r)
      omch[wave][(r + 8 * half) * 16 + m] = d[r] + bias;
    __syncthreads();

    // ---- bilinear sampling: 64 tasks (16 px x 4 taps), 2 per lane ---------
#pragma unroll
    for (int tt = 0; tt < 2; ++tt) {
      const int task = lane + 32 * tt;
      const int tm = task & 15, tj = task >> 4;
      const int pidx = ch * 4 + tj;
      float s = 0.0f;
      if (pidx < CK) {
        const float offy = omch[wave][tm * 16 + 3 * tj];
        const float offx = omch[wave][tm * 16 + 3 * tj + 1];
        float msk        = omch[wave][tm * 16 + 3 * tj + 2];
        msk = 1.0f / (1.0f + __expf(-msk));
        const int c  = pidx / DS, kk = pidx - c * DS;
        const int ky = kk / KS,   kx = kk - ky * KS;
        const float py = offy + (float)(ky + y - PAD);
        const float px = offx + (float)(kx + x0 + tm - PAD);
        const float fy = floorf(py), fx = floorf(px);
        const float wy = py - fy,    wx = px - fx;
        const int y0i = (int)fy, x0i = (int)fx;
        const float* ip = inp + (size_t)(b * CIN + c) * HWSZ;
        float v00 = 0.f, v01 = 0.f, v10 = 0.f, v11 = 0.f;
        const bool yok0 = (y0i >= 0) & (y0i < HH);
        const bool yok1 = (y0i + 1 >= 0) & (y0i + 1 < HH);
        const bool xok0 = (x0i >= 0) & (x0i < WW);
        const bool xok1 = (x0i + 1 >= 0) & (x0i + 1 < WW);
        if (yok0 && xok0) v00 = ip[y0i * WW + x0i];
        if (yok0 && xok1) v01 = ip[y0i * WW + x0i + 1];
        if (yok1 && xok0) v10 = ip[(y0i + 1) * WW + x0i];
        if (yok1 && xok1) v11 = ip[(y0i + 1) * WW + x0i + 1];
        s = (v00 * (1.f - wy) * (1.f - wx) + v01 * (1.f - wy) * wx +
             v10 * wy * (1.f - wx)         + v11 * wy * wx) * msk;
      }
      sampb[wave][tj * 16 + tm] = s;
    }
    __syncthreads();

    // ---- dcn GEMM step: D(16x64) += samp(16x4) * dcn_w^T(4x64) ------------
    v2f a;
    a.x = sampb[wave][kb * 16 + m];
    a.y = sampb[wave][(kb + 1) * 16 + m];
    const int k0 = ch * 4 + kb;
#pragma unroll
    for (int oT = 0; oT < 4; ++oT) {
      const int o = oT * 16 + m;
      v2f bb;
      bb.x = (k0     < CK) ? dcnw[o * CK + k0]     : 0.0f;
      bb.y = (k0 + 1 < CK) ? dcnw[o * CK + k0 + 1] : 0.0f;
      acc[oT] = wmma4(a, bb, acc[oT]);
    }
  }

  // ---- epilogue: bias + relu + store ---------------------------------------
#pragma unroll
  for (int oT = 0; oT < 4; ++oT) {
    const int o = oT * 16 + m;
    const float bias = dcnb[o];
#pragma unroll
    for (int r = 0; r < 8; ++r) {
      const int mm = r + 8 * half;
      float v = acc[oT][r] + bias;
      v = v > 0.f ? v : 0.f;
      outw[(size_t)(b * OC + o) * HWSZ + rem + mm] = v;
    }
  }
}

// ---------------------------------------------------------------------------
// Phase 2a: zero the attention accumulator
// ---------------------------------------------------------------------------
__global__ void zero_accum_kernel(float* __restrict__ accum) {
  accum[threadIdx.x] = 0.0f;   // 256 = BATCH*OC entries, one block of 256
}

// ---------------------------------------------------------------------------
// Phase 2b: attn = relu(W_attn * outcat + b), summed over pixels (mean later)
// ---------------------------------------------------------------------------
__global__ __launch_bounds__(256) void attn_pool_kernel(
    const float* __restrict__ outw,   // [3][BATCH][64][HWSZ]
    const float* __restrict__ attw, const float* __restrict__ attb,
    float* __restrict__ accum)        // [BATCH][64]
{
  const int wave = threadIdx.x >> 5, lane = threadIdx.x & 31;
  const int tile = blockIdx.x * 8 + wave;
  const int b    = tile / (HWSZ / 16);
  const int p2   = (tile - b * (HWSZ / 16)) * 16;
  const int m = lane & 15, half = lane >> 4, kb = half * 2;

  v8f acc[4] = {};
  for (int t = 0; t < 48; ++t) {
    const int k0 = 4 * t + kb;
    v2f a;
    a.x = outw[((size_t)((k0 >> 6) * BATCH + b) * OC + (k0 & 63)) * HWSZ + p2 + m];
    a.y = outw[((size_t)(((k0 + 1) >> 6) * BATCH + b) * OC + ((k0 + 1) & 63)) * HWSZ + p2 + m];
#pragma unroll
    for (int oT = 0; oT < 4; ++oT) {
      const int o = oT * 16 + m;
      v2f bb;
      bb.x = attw[o * 192 + k0];
      bb.y = attw[o * 192 + k0 + 1];
      acc[oT] = wmma4(a, bb, acc[oT]);
    }
  }
#pragma unroll
  for (int oT = 0; oT < 4; ++oT) {
    const int o = oT * 16 + m;
    const float bias = attb[o];
    float s = 0.f;
#pragma unroll
    for (int r = 0; r < 8; ++r) {
      float v = acc[oT][r] + bias;
      s += v > 0.f ? v : 0.f;
    }
    s += __shfl_xor(s, 16);               // fold the two lane-halves (wave32)
    if (half == 0) atomicAdd(&accum[b * OC + o], s);
  }
}

// ---------------------------------------------------------------------------
// Phase 2c: mean -> fc(relu) -> per-scale fcs -> atts[b][s][o]
// ---------------------------------------------------------------------------
__global__ __launch_bounds__(256) void attn_tail_kernel(
    const float* __restrict__ accum,
    const float* __restrict__ fcw, const float* __restrict__ fcb,
    const float* __restrict__ fw0, const float* __restrict__ fb0,
    const float* __restrict__ fw1, const float* __restrict__ fb1,
    const float* __restrict__ fw2, const float* __restrict__ fb2,
    float* __restrict__ atts)             // [BATCH][3][64]
{
  __shared__ float am[BATCH * OC];
  __shared__ float at[BATCH * LC];
  const int t = threadIdx.x;
  am[t] = accum[t] * (1.0f / (float)HWSZ);
  __syncthreads();
  if (t < BATCH * LC) {
    const int b = t >> 5, l = t & 31;
    float v = fcb[l];
    for (int o = 0; o < OC; ++o) v += fcw[l * OC + o] * am[b * OC + o];
    at[b * LC + l] = v > 0.f ? v : 0.f;
  }
  __syncthreads();
  for (int idx = t; idx < BATCH * 3 * OC; idx += 256) {
    const int b = idx / 192;
    const int r = idx - b * 192;
    const int s = r >> 6, o = r & 63;
    const float* w  = (s == 0) ? fw0 : (s == 1) ? fw1 : fw2;
    const float* bb = (s == 0) ? fb0 : (s == 1) ? fb1 : fb2;
    float v = bb[o];
    for (int l = 0; l < LC; ++l) v += w[o * LC + l] * at[b * LC + l];
    atts[idx] = v;
  }
}

// ---------------------------------------------------------------------------
// Phase 3: out = relu(conv_w * (outcat * atts) + conv_b)
// ---------------------------------------------------------------------------
__global__ __launch_bounds__(256) void final_conv_kernel(
    const float* __restrict__ outw, const float* __restrict__ atts,
    const float* __restrict__ convw, const float* __restrict__ convb,
    float* __restrict__ out)
{
  const int b = blockIdx.y;
  __shared__ float sc[192];
  if (threadIdx.x < 192) sc[threadIdx.x] = atts[b * 192 + threadIdx.x];
  __syncthreads();

  const int wave = threadIdx.x >> 5, lane = threadIdx.x & 31;
  const int tile = blockIdx.x * 8 + wave;
  const int p2 = tile * 16;
  const int m = lane & 15, half = lane >> 4, kb = half * 2;

  v8f acc[4] = {};
  for (int t = 0; t < 48; ++t) {
    const int k0 = 4 * t + kb;
    v2f a;
    a.x = outw[((size_t)((k0 >> 6) * BATCH + b) * OC + (k0 & 63)) * HWSZ + p2 + m] * sc[k0];
    a.y = outw[((size_t)(((k0 + 1) >> 6) * BATCH + b) * OC + ((k0 + 1) & 63)) * HWSZ + p2 + m] * sc[k0 + 1];
#pragma unroll
    for (int oT = 0; oT < 4; ++oT) {
      const int o = oT * 16 + m;
      v2f bb;
      bb.x = convw[o * 192 + k0];
      bb.y = convw[o * 192 + k0 + 1];
      acc[oT] = wmma4(a, bb, acc[oT]);
    }
  }
#pragma unroll
  for (int oT = 0; oT < 4; ++oT) {
    const int o = oT * 16 + m;
    const float bias = convb[o];
#pragma unroll
    for (int r = 0; r < 8; ++r) {
      const int mm = r + 8 * half;
      float v = acc[oT][r] + bias;
      v = v > 0.f ? v : 0.f;
      out[((size_t)(b * OC + o)) * HWSZ + p2 + mm] = v;
    }
  }
}

// ---------------------------------------------------------------------------
extern "C" void kernel_launch(void* const* d_in, const int* in_sizes, int n_in,
                              void* d_out, int out_size, void* d_ws, size_t ws_size,
                              hipStream_t stream) {
  const float* fea = (const float*)d_in[0];
  const float* inp = (const float*)d_in[1];
  const float* attw = (const float*)d_in[20];
  const float* attb = (const float*)d_in[21];
  const float* fcw  = (const float*)d_in[22];
  const float* fcb  = (const float*)d_in[23];
  const float* fw0  = (const float*)d_in[24];
  const float* fb0  = (const float*)d_in[25];
  const float* fw1  = (const float*)d_in[26];
  const float* fb1  = (const float*)d_in[27];
  const float* fw2  = (const float*)d_in[28];
  const float* fb2  = (const float*)d_in[29];
  const float* cw   = (const float*)d_in[30];
  const float* cb   = (const float*)d_in[31];

  float* ws = (float*)d_ws;
  const size_t SCALE_SZ = (size_t)BATCH * OC * HWSZ;   // per-scale stacked out
  float* outw  = ws;                                   // 3 * SCALE_SZ floats
  float* accum = ws + 3 * SCALE_SZ;                    // BATCH*OC
  float* atts  = accum + BATCH * OC;                   // BATCH*3*OC

  const int nTiles  = BATCH * HWSZ / 16;               // 9216, exact
  const int nBlocks = nTiles / 8;                      // 1152, exact

  scale_kernel<1><<<nBlocks, 256, 0, stream>>>(
      fea, inp, (const float*)d_in[2], (const float*)d_in[3],
      (const float*)d_in[4], (const float*)d_in[5],
      (const float*)d_in[6], (const float*)d_in[7], outw + 0 * SCALE_SZ);
  scale_kernel<3><<<nBlocks, 256, 0, stream>>>(
      fea, inp, (const float*)d_in[8], (const float*)d_in[9],
      (const float*)d_in[10], (const float*)d_in[11],
      (const float*)d_in[12], (const float*)d_in[13], outw + 1 * SCALE_SZ);
  scale_kernel<5><<<nBlocks, 256, 0, stream>>>(
      fea, inp, (const float*)d_in[14], (const float*)d_in[15],
      (const float*)d_in[16], (const float*)d_in[17],
      (const float*)d_in[18], (const float*)d_in[19], outw + 2 * SCALE_SZ);

  zero_accum_kernel<<<1, BATCH * OC, 0, stream>>>(accum);
  attn_pool_kernel<<<nBlocks, 256, 0, stream>>>(outw, attw, attb, accum);
  attn_tail_kernel<<<1, 256, 0, stream>>>(accum, fcw, fcb, fw0, fb0, fw1, fb1,
                                          fw2, fb2, atts);

  dim3 g((HWSZ / 16) / 8, BATCH);                      // 288 x 4, exact
  final_conv_kernel<<<g, 256, 0, stream>>>(outw, atts, cw, cb, (float*)d_out);
}